// WHVILinear_29377576305183
// MI455X (gfx1250) — compile-verified
//
#include <hip/hip_runtime.h>
#include <math.h>

// WHVI fused kernel for MI455X (gfx1250, wave32).
// out = s1 * WHT( u * WHT( s2 * x ) ) / 2048,  u = g_mu + sqrt(softplus(g_rho))*eps
// WHT_2048 = H16 (bits0-3, WMMA f32) x H16 (bits4-7, WMMA f32) x H8 (bits8-10, VALU).
// All LDS addresses decompose into per-lane base + compile-time immediate offsets.

#define WHVI_D      2048
#define ROWS_PER_WG 8
#define ROW_LDS     2176   // 2048 floats + 4-float pad per 64 floats (bank-conflict-free, 16B aligned)

typedef __attribute__((ext_vector_type(2))) float v2f;
typedef __attribute__((ext_vector_type(4))) float v4f;
typedef __attribute__((ext_vector_type(8))) float v8f;

static __device__ __forceinline__ float hsign(int a, int b) {
    // Sylvester Hadamard entry: (-1)^popcount(a&b)
    return (__builtin_popcount(a & b) & 1) ? -1.0f : 1.0f;
}

// Radix-16 x radix-16 stages (bits 0-3 then bits 4-7) of the 2048-pt WHT, in-place
// on one padded LDS row, one wave32, exact fp32 via V_WMMA_F32_16X16X4_F32.
// Padded index: idx(j) = j + 4*(j>>6). For every access pattern below this splits
// carry-free into (per-lane constant) + (compile-time constant), so after full
// unrolling all DS ops are base-register + immediate-offset.
static __device__ __forceinline__ void fwht_stages12(float* row, int lane, const v2f* sg) {
    const int mN    = lane & 15;         // M (A row) or N (B/D column) index
    const int kh    = (lane >> 4) << 1;  // K offset from lane half
    const int mbase = (lane >> 4) << 3;  // D-matrix M offset from lane half

    // ---- Stage 1: radix-16 along bits 0-3.  D = A(data tile) x B(H16 signs)
    {
        const float* pA = row + mN * 16 + ((mN >> 2) << 2) + kh;          // + t*272 + 4q (+v)
        float*       pD = row + mN + mbase * 16 + ((mbase >> 2) << 2);    // + t*272 + v*16 + (v>>2)*4
        #pragma unroll
        for (int t = 0; t < 8; ++t) {
            v8f c = {};
            #pragma unroll
            for (int q = 0; q < 4; ++q) {
                v2f a;
                a.x = pA[t * 272 + 4 * q];
                a.y = pA[t * 272 + 4 * q + 1];
                c = __builtin_amdgcn_wmma_f32_16x16x4_f32(false, a, false, sg[q],
                                                          (short)0, c, false, false);
            }
            #pragma unroll
            for (int v = 0; v < 8; ++v)
                pD[t * 272 + v * 16 + ((v >> 2) << 2)] = c[v];
        }
    }

    // ---- Stage 2: radix-16 along bits 4-7.  D = A(H16 signs) x B(data)
    {
        const float* pB = row + mN + kh * 16;                             // + h*272 + q*68 (+v*16)
        float*       pD = row + mN + mbase * 16 + ((mbase >> 2) << 2);    // + h*272 + v*16 + (v>>2)*4
        #pragma unroll
        for (int h = 0; h < 8; ++h) {
            v8f c = {};
            #pragma unroll
            for (int q = 0; q < 4; ++q) {
                v2f b;
                b.x = pB[h * 272 + q * 68];
                b.y = pB[h * 272 + q * 68 + 16];
                c = __builtin_amdgcn_wmma_f32_16x16x4_f32(false, sg[q], false, b,
                                                          (short)0, c, false, false);
            }
            #pragma unroll
            for (int v = 0; v < 8; ++v)
                pD[h * 272 + v * 16 + ((v >> 2) << 2)] = c[v];
        }
    }
}

// Radix-8 butterfly on v[0..7] (float4 lanes), Sylvester order.
static __device__ __forceinline__ void radix8(v4f v[8]) {
    #pragma unroll
    for (int s = 4; s >= 1; s >>= 1) {
        #pragma unroll
        for (int i = 0; i < 8; ++i) {
            if ((i & s) == 0) {
                const v4f lo = v[i], hi = v[i + s];
                v[i]     = lo + hi;
                v[i + s] = lo - hi;
            }
        }
    }
}

__global__ void whvi_prepare_u(const float* __restrict__ g_mu, const float* __restrict__ g_rho,
                               const float* __restrict__ eps, float* __restrict__ u, int n) {
    const int j = blockIdx.x * blockDim.x + threadIdx.x;
    if (j < n) {
        const float r  = g_rho[j];
        const float sp = fmaxf(r, 0.0f) + log1pf(expf(-fabsf(r)));  // stable softplus
        u[j] = g_mu[j] + sqrtf(sp) * eps[j];
    }
}

__launch_bounds__(256)
__global__ void whvi_main(const float* __restrict__ x, const float* __restrict__ s1,
                          const float* __restrict__ s2, const float* __restrict__ u,
                          float* __restrict__ out) {
    __shared__ float lds[ROWS_PER_WG * ROW_LDS];

    const int lane = threadIdx.x & 31;
    const int wave = threadIdx.x >> 5;
    float* row = &lds[wave * ROW_LDS];

    const long long r = (long long)blockIdx.x * ROWS_PER_WG + wave;
    const float* xr   = x + r * WHVI_D;
    float*       outr = out + r * WHVI_D;

    // Per-lane H16 fragment signs (stage-1 B-matrix == stage-2 A-matrix; hsign symmetric).
    const int mN = lane & 15;
    const int kh = (lane >> 4) << 1;
    v2f sg[4];
    #pragma unroll
    for (int q = 0; q < 4; ++q) {
        sg[q].x = hsign(4 * q + kh, mN);
        sg[q].y = hsign(4 * q + kh + 1, mN);
    }

    // Phase 0: coalesced 128-bit loads of x, fused *s2, into padded LDS row.
    {
        float* pl = row + 4 * lane + ((lane >> 4) << 2);   // + it*136
        #pragma unroll
        for (int it = 0; it < 16; ++it) {
            const int j = it * 128 + lane * 4;
            v4f xv = *(const v4f*)(xr + j);
            xv *= *(const v4f*)(s2 + j);
            *(v4f*)(pl + it * 136) = xv;
        }
    }

    // ===== Transform 1 =====
    fwht_stages12(row, lane, sg);
    // Stage 3 (radix-8, stride 256) fused with *u, back to LDS.
    {
        #pragma unroll
        for (int it = 0; it < 2; ++it) {
            const int j0 = it * 128 + lane * 4;
            float* p = row + j0 + ((j0 >> 6) << 2);        // + 272*k
            v4f v[8];
            #pragma unroll
            for (int k = 0; k < 8; ++k) v[k] = *(const v4f*)(p + 272 * k);
            radix8(v);
            #pragma unroll
            for (int k = 0; k < 8; ++k) {
                v[k] *= *(const v4f*)(u + j0 + 256 * k);
                *(v4f*)(p + 272 * k) = v[k];
            }
        }
    }

    // ===== Transform 2 =====
    fwht_stages12(row, lane, sg);
    // Stage 3 fused with *s1 and 1/2048 normalization, stored straight to global
    // (lane*4 contiguous across the wave -> 512B coalesced stores).
    {
        const float invD = 1.0f / (float)WHVI_D;
        #pragma unroll
        for (int it = 0; it < 2; ++it) {
            const int j0 = it * 128 + lane * 4;
            const float* p = row + j0 + ((j0 >> 6) << 2);  // + 272*k
            v4f v[8];
            #pragma unroll
            for (int k = 0; k < 8; ++k) v[k] = *(const v4f*)(p + 272 * k);
            radix8(v);
            #pragma unroll
            for (int k = 0; k < 8; ++k) {
                v4f val = v[k] * *(const v4f*)(s1 + j0 + 256 * k);
                val *= invD;
                *(v4f*)(outr + j0 + 256 * k) = val;
            }
        }
    }
}

extern "C" void kernel_launch(void* const* d_in, const int* in_sizes, int n_in,
                              void* d_out, int out_size, void* d_ws, size_t ws_size,
                              hipStream_t stream) {
    const float* x     = (const float*)d_in[0];
    const float* s1    = (const float*)d_in[1];
    const float* s2    = (const float*)d_in[2];
    const float* g_mu  = (const float*)d_in[3];
    const float* g_rho = (const float*)d_in[4];
    const float* eps   = (const float*)d_in[5];
    // d_in[6] (H) intentionally unused: H is reconstructed structurally.

    float* u = (float*)d_ws;  // WHVI_D floats of scratch

    whvi_prepare_u<<<(WHVI_D + 255) / 256, 256, 0, stream>>>(g_mu, g_rho, eps, u, WHVI_D);

    const int rows = in_sizes[0] / WHVI_D;   // 8192
    whvi_main<<<rows / ROWS_PER_WG, 256, 0, stream>>>(x, s1, s2, u, (float*)d_out);
}